// AdAP_PZ_52587579572535
// MI455X (gfx1250) — compile-verified
//
#include <hip/hip_runtime.h>

// ---------------------------------------------------------------------------
// AdAP loss, single scalar output.
//   nat part:  N x N squared-hinge row reductions done with f32 WMMA
//              (V_WMMA_F32_16X16X4_F32). One WMMA per 4 columns computes BOTH
//              rowsum (B column 0 = ones) and possum (B column 1 = posf).
//   adv part:  O(N) elementwise KL-ish term + n_pos count.
//   combine:   fixed-order tree reductions (deterministic, no atomics).
// ---------------------------------------------------------------------------

#define NN      12288
#define GAMMA   0.1f
#define LAMBDA  1.0f
#define EPSF    1e-12f
#define JSPLIT  4
#define JCHUNK  (NN / JSPLIT)   // 3072

typedef __attribute__((ext_vector_type(2))) float v2f;
typedef __attribute__((ext_vector_type(8))) float v8f;

// --------------------------------------------------------------------------
// Kernel 1: pairwise hinge^2 row sums.
// grid = (192, JSPLIT), block = 128 (4 waves). Each wave owns 16 rows,
// each block owns 64 rows x one j-chunk of 3072 columns.
//
// WMMA layout (32-bit A 16x4 / B 4x16, ISA 7.12.2):
//   A: lanes 0-15 hold rows M=0..15 at K=0 (v0) / K=1 (v1);
//      lanes 16-31 hold K=2 / K=3.
//   B: lane n&15 holds column n; VGPR0 = K=0 (lanes<16) / K=2 (lanes>=16),
//      VGPR1 = K=1 / K=3.
// We set B[:,0]=1, B[:,1]=posf, B[:,2..15]=0, so a single accumulator D
// carries rowsum in column 0 and possum in column 1.
// Row sums are order-agnostic over j, so per 8-column step lanes 0-15 take
// j0..j0+3 and lanes 16-31 take j0+4..j0+7 -> one b128 LDS load per operand.
// D layout: VGPR r, lanes 0-15 -> (M=r, N=lane); lanes 16-31 -> (M=8+r).
// --------------------------------------------------------------------------
__global__ __launch_bounds__(128) void pairwise_kernel(
    const float* __restrict__ y_pred,
    const int*   __restrict__ y_true,
    float* __restrict__ rowsumPart,
    float* __restrict__ possumPart)
{
    __shared__ __align__(16) float lds_f[NN];    // 48 KB
    __shared__ __align__(16) float lds_pf[NN];   // 48 KB (posf as f32)
    __shared__ float xr[4][16];
    __shared__ float xp[4][16];

    for (int t = threadIdx.x; t < NN / 4; t += 128) {
        ((float4*)lds_f)[t] = ((const float4*)y_pred)[t];
        int4 yt = ((const int4*)y_true)[t];
        float4 pf;
        pf.x = (yt.x == 1) ? 1.0f : 0.0f;
        pf.y = (yt.y == 1) ? 1.0f : 0.0f;
        pf.z = (yt.z == 1) ? 1.0f : 0.0f;
        pf.w = (yt.w == 1) ? 1.0f : 0.0f;
        ((float4*)lds_pf)[t] = pf;
    }
    __syncthreads();

    const int   lane    = threadIdx.x & 31;
    const int   wave    = threadIdx.x >> 5;
    const int   rowBase = blockIdx.x * 64 + wave * 16;
    const float fi      = lds_f[rowBase + (lane & 15)];
    const float ai      = 1.0f - fi;             // MARGIN - f[i]
    const int   n       = lane & 15;             // D/B column owned by lane
    const float c0      = (n == 0) ? 1.0f : 0.0f;   // rowsum column select
    const float cm      = (n == 1) ? 1.0f : 0.0f;   // possum column select
    const int   jsel    = (lane >> 4) << 2;      // 0 for lanes 0-15, 4 for 16-31
    const int   jbeg    = blockIdx.y * JCHUNK;
    const int   jend    = jbeg + JCHUNK;

    v8f acc = {};

#pragma unroll 4
    for (int j0 = jbeg; j0 < jend; j0 += 8) {
        const int jb = (j0 + jsel) >> 2;         // float4 index, 16B aligned
        float4 f4 = ((const float4*)lds_f)[jb];
        float4 p4 = ((const float4*)lds_pf)[jb];

        float t0 = fmaxf(ai + f4.x, 0.0f);
        float t1 = fmaxf(ai + f4.y, 0.0f);
        float t2 = fmaxf(ai + f4.z, 0.0f);
        float t3 = fmaxf(ai + f4.w, 0.0f);

        v2f A0 = { t0 * t0, t1 * t1 };
        v2f B0 = { fmaf(cm, p4.x, c0), fmaf(cm, p4.y, c0) };
        acc = __builtin_amdgcn_wmma_f32_16x16x4_f32(
                  false, A0, false, B0, (short)0, acc, false, false);

        v2f A1 = { t2 * t2, t3 * t3 };
        v2f B1 = { fmaf(cm, p4.z, c0), fmaf(cm, p4.w, c0) };
        acc = __builtin_amdgcn_wmma_f32_16x16x4_f32(
                  false, A1, false, B1, (short)0, acc, false, false);
    }

    // Extract: column 0 (lanes 0/16) = rowsum, column 1 (lanes 1/17) = possum.
    if (lane == 0) {
#pragma unroll
        for (int r = 0; r < 8; ++r) xr[wave][r] = acc[r];
    }
    if (lane == 16) {
#pragma unroll
        for (int r = 0; r < 8; ++r) xr[wave][8 + r] = acc[r];
    }
    if (lane == 1) {
#pragma unroll
        for (int r = 0; r < 8; ++r) xp[wave][r] = acc[r];
    }
    if (lane == 17) {
#pragma unroll
        for (int r = 0; r < 8; ++r) xp[wave][8 + r] = acc[r];
    }
    __syncthreads();

    if (lane < 16) {
        int row = rowBase + lane;
        rowsumPart[blockIdx.y * NN + row] = xr[wave][lane];
        possumPart[blockIdx.y * NN + row] = xp[wave][lane];
    }
}

// --------------------------------------------------------------------------
// Kernel 2: per-row AdAP coefficient + deterministic block reduction.
// nat contribution for positive rows:
//   (up_new*rowsum - ua_new*possum) / ua_new^2
// --------------------------------------------------------------------------
__global__ __launch_bounds__(256) void row_finalize_kernel(
    const float* __restrict__ rowsumPart,
    const float* __restrict__ possumPart,
    const int*   __restrict__ y_true,
    const int*   __restrict__ index_s,
    const float* __restrict__ u_all,
    const float* __restrict__ u_pos,
    float* __restrict__ natPartials)
{
    __shared__ float red[256];
    int i = blockIdx.x * 256 + threadIdx.x;

    float rowsum = 0.0f, possum = 0.0f;
#pragma unroll
    for (int c = 0; c < JSPLIT; ++c) {
        rowsum += rowsumPart[c * NN + i];
        possum += possumPart[c * NN + i];
    }

    float contrib = 0.0f;
    if (y_true[i] == 1) {
        int   idx    = index_s[i];
        float ua     = u_all[idx];
        float up     = u_pos[idx];
        const float invN = 1.0f / (float)NN;
        float ua_new = (1.0f - GAMMA) * ua + GAMMA * (rowsum * invN);
        float up_new = (1.0f - GAMMA) * up + GAMMA * (possum * invN);
        float denom  = ua_new * ua_new;
        contrib = (up_new * rowsum - ua_new * possum) / denom;
    }
    red[threadIdx.x] = contrib;
    __syncthreads();
    for (int s = 128; s > 0; s >>= 1) {
        if (threadIdx.x < s) red[threadIdx.x] += red[threadIdx.x + s];
        __syncthreads();
    }
    if (threadIdx.x == 0) natPartials[blockIdx.x] = red[0];
}

// --------------------------------------------------------------------------
// Kernel 3: adversarial KL-style term + n_pos count (O(N) elementwise).
// --------------------------------------------------------------------------
__global__ __launch_bounds__(256) void adv_kernel(
    const float* __restrict__ y_pred,
    const float* __restrict__ y_pred_adv,
    const int*   __restrict__ y_true,
    float* __restrict__ advPartials,
    float* __restrict__ nposPartials)
{
    __shared__ float sa[256];
    __shared__ float sp[256];
    int i = blockIdx.x * 256 + threadIdx.x;

    float yp  = y_pred[i];
    float ypa = y_pred_adv[i];
    float q0 = yp, q1 = 1.0f - yp;
    float x0 = (q0 > 0.0f) ? q0 * logf(q0) : 0.0f;   // xlogy(p,p)
    float x1 = (q1 > 0.0f) ? q1 * logf(q1) : 0.0f;
    float adv = x0 + x1 - q0 * logf(ypa + EPSF)
                        - q1 * logf((1.0f - ypa) + EPSF);

    sa[threadIdx.x] = adv;
    sp[threadIdx.x] = (y_true[i] == 1) ? 1.0f : 0.0f;
    __syncthreads();
    for (int s = 128; s > 0; s >>= 1) {
        if (threadIdx.x < s) {
            sa[threadIdx.x] += sa[threadIdx.x + s];
            sp[threadIdx.x] += sp[threadIdx.x + s];
        }
        __syncthreads();
    }
    if (threadIdx.x == 0) {
        advPartials[blockIdx.x]  = sa[0];
        nposPartials[blockIdx.x] = sp[0];
    }
}

// --------------------------------------------------------------------------
// Kernel 4: final scalar combine (48 partials each).
// --------------------------------------------------------------------------
__global__ __launch_bounds__(64) void finalize_kernel(
    const float* __restrict__ natPartials,
    const float* __restrict__ advPartials,
    const float* __restrict__ nposPartials,
    float* __restrict__ out)
{
    __shared__ float sn[64], sa[64], sp[64];
    int t = threadIdx.x;
    sn[t] = (t < 48) ? natPartials[t]  : 0.0f;
    sa[t] = (t < 48) ? advPartials[t]  : 0.0f;
    sp[t] = (t < 48) ? nposPartials[t] : 0.0f;
    __syncthreads();
    for (int s = 32; s > 0; s >>= 1) {
        if (t < s) { sn[t] += sn[t + s]; sa[t] += sa[t + s]; sp[t] += sp[t + s]; }
        __syncthreads();
    }
    if (t == 0) {
        float n_pos = sp[0];
        float nat   = sn[0] / (n_pos * (float)NN);
        out[0] = nat + LAMBDA * (sa[0] / (float)NN);
    }
}

// --------------------------------------------------------------------------
extern "C" void kernel_launch(void* const* d_in, const int* in_sizes, int n_in,
                              void* d_out, int out_size, void* d_ws, size_t ws_size,
                              hipStream_t stream)
{
    const float* y_pred     = (const float*)d_in[0];
    const float* y_pred_adv = (const float*)d_in[1];
    const float* u_all      = (const float*)d_in[2];
    const float* u_pos      = (const float*)d_in[3];
    const int*   y_true     = (const int*)d_in[4];
    const int*   index_s    = (const int*)d_in[5];

    float* ws_f       = (float*)d_ws;
    float* rowsumPart = ws_f;                       // JSPLIT*NN
    float* possumPart = ws_f + JSPLIT * NN;         // JSPLIT*NN
    float* natP       = ws_f + 2 * JSPLIT * NN;     // 48
    float* advP       = natP + 48;                  // 48
    float* nposP      = advP + 48;                  // 48

    dim3 gridA(NN / 64, JSPLIT);
    pairwise_kernel<<<gridA, 128, 0, stream>>>(y_pred, y_true,
                                               rowsumPart, possumPart);
    row_finalize_kernel<<<NN / 256, 256, 0, stream>>>(rowsumPart, possumPart,
                                                      y_true, index_s,
                                                      u_all, u_pos, natP);
    adv_kernel<<<NN / 256, 256, 0, stream>>>(y_pred, y_pred_adv, y_true,
                                             advP, nposP);
    finalize_kernel<<<1, 64, 0, stream>>>(natP, advP, nposP, (float*)d_out);
}